// GatedSwitchGNN_globalMLP_28776280883955
// MI455X (gfx1250) — compile-verified
//
#include <hip/hip_runtime.h>
#include <math.h>

typedef __attribute__((ext_vector_type(2))) float v2f;
typedef __attribute__((ext_vector_type(8))) float v8f;

#define B_    200
#define V_    40
#define FIN_  16
#define H_    128
#define NSW_  12
#define NE_   40
#define M_    52
#define HID_  1024
#define OUT_  104   // M_ + V_ + NSW_

// ---------------------------------------------------------------------------
// f32 WMMA GEMM:  C[M,N] = act(A[M,K] @ B[K,N] + bias)
// Block = 128 threads = 4 waves; each wave owns one 16-wide N tile
// (ntile = blockIdx.x*4 + wave) and computes a 64(M)x16(N) strip as
// 4 register-blocked 16x16 accumulators -> each B operand feeds 4 WMMAs.
// Interior tiles take an unguarded pointer-bumping fast loop; only edge
// tiles (M=200 tail, N=104 tail) take the guarded path.
//   V_WMMA_F32_16X16X4_F32 operand layout (wave32):
//     A (16x4): lanes 0-15 -> M rows with K={0,1}; lanes 16-31 -> K={2,3}
//     B (4x16): lanes 0-15 -> N cols with K={0,1}; lanes 16-31 -> K={2,3}
//     C/D     : VGPR g: lanes 0-15 -> M=g, lanes 16-31 -> M=g+8
// K must be a multiple of 4 (16 / 128 / 1024 / 6656 all qualify).
// ---------------------------------------------------------------------------
__global__ void wmma_gemm_f32(const float* __restrict__ A, const float* __restrict__ Bm,
                              float* __restrict__ C, const float* __restrict__ bias,
                              int M, int N, int K, int do_relu)
{
    const int lane  = threadIdx.x & 31;
    const int wave  = threadIdx.x >> 5;
    const int lo    = lane & 15;
    const int hi    = lane >> 4;              // 0 -> K pair {0,1}, 1 -> {2,3}
    const int ntile = blockIdx.x * 4 + wave;
    const int mbase = blockIdx.y * 64;
    if (ntile * 16 >= N) return;              // uniform per wave

    const int col = ntile * 16 + lo;
    v8f acc0 = {}, acc1 = {}, acc2 = {}, acc3 = {};

    const bool full = (mbase + 64 <= M) && (ntile * 16 + 16 <= N);

    if (full) {
        // -------- fast path: no guards, pointer-bumping loop --------
        const float* ap = A + (size_t)(mbase + lo) * K + 2 * hi;
        const float* bp = Bm + (size_t)(2 * hi) * N + col;
        const size_t arstride = (size_t)16 * K;
        for (int k = 0; k < K; k += 4) {
            v2f a0 = *(const v2f*)(ap);
            v2f a1 = *(const v2f*)(ap + arstride);
            v2f a2 = *(const v2f*)(ap + 2 * arstride);
            v2f a3 = *(const v2f*)(ap + 3 * arstride);
            v2f b;  b.x = bp[0];  b.y = bp[(size_t)N];
            acc0 = __builtin_amdgcn_wmma_f32_16x16x4_f32(false, a0, false, b, (short)0, acc0, false, false);
            acc1 = __builtin_amdgcn_wmma_f32_16x16x4_f32(false, a1, false, b, (short)0, acc1, false, false);
            acc2 = __builtin_amdgcn_wmma_f32_16x16x4_f32(false, a2, false, b, (short)0, acc2, false, false);
            acc3 = __builtin_amdgcn_wmma_f32_16x16x4_f32(false, a3, false, b, (short)0, acc3, false, false);
            ap += 4;
            bp += (size_t)4 * N;
        }
        const float bv = bias ? bias[col] : 0.f;
#pragma unroll
        for (int g = 0; g < 8; ++g) {
            const int r = mbase + g + 8 * hi;
            float v0 = acc0[g] + bv, v1 = acc1[g] + bv, v2 = acc2[g] + bv, v3 = acc3[g] + bv;
            if (do_relu) {
                v0 = fmaxf(v0, 0.f); v1 = fmaxf(v1, 0.f);
                v2 = fmaxf(v2, 0.f); v3 = fmaxf(v3, 0.f);
            }
            C[(size_t)(r)      * N + col] = v0;
            C[(size_t)(r + 16) * N + col] = v1;
            C[(size_t)(r + 32) * N + col] = v2;
            C[(size_t)(r + 48) * N + col] = v3;
        }
    } else {
        // -------- guarded path: edge tiles only (rare) --------
        v8f acc[4] = {acc0, acc1, acc2, acc3};
#pragma unroll
        for (int sub = 0; sub < 4; ++sub) {
            const int arow = mbase + sub * 16 + lo;
            v8f a8 = {};
            for (int k = 0; k < K; k += 4) {
                const int ka = k + 2 * hi;
                v2f a; a.x = 0.f; a.y = 0.f;
                if (arow < M) a = *(const v2f*)(A + (size_t)arow * K + ka);
                v2f b; b.x = 0.f; b.y = 0.f;
                if (col < N) {
                    b.x = Bm[(size_t)ka * N + col];
                    b.y = Bm[(size_t)(ka + 1) * N + col];
                }
                a8 = __builtin_amdgcn_wmma_f32_16x16x4_f32(false, a, false, b, (short)0, a8, false, false);
            }
            acc[sub] = a8;
        }
        const float bv = (bias && col < N) ? bias[col] : 0.f;
#pragma unroll
        for (int sub = 0; sub < 4; ++sub) {
#pragma unroll
            for (int g = 0; g < 8; ++g) {
                const int row = mbase + sub * 16 + g + 8 * hi;
                if (row < M && col < N) {
                    float v = acc[sub][g] + bv;
                    if (do_relu) v = fmaxf(v, 0.f);
                    C[(size_t)row * N + col] = v;
                }
            }
        }
    }
}

// ---------------------------------------------------------------------------
// Layer-0 fusion. One block per batch, 128 threads (= H columns).
// e0[b,k,h] = (embed[1]@WA0)[h] + xB[b,i_k,h] + xC[b,j_k,h]  (switch pairs only)
// s1 = relu(e0);  gates0 = sigmoid(e0)
// x1[b,i,h] = relu(Ux[b,i,h] + sum_edges Vx + sum_sw gate*Vx)
// Each thread owns one h column across all 40 nodes -> race-free LDS scatter.
// ---------------------------------------------------------------------------
__global__ void layer0_post(const float* __restrict__ Ux, const float* __restrict__ Vx,
                            const float* __restrict__ xB, const float* __restrict__ xC,
                            const float* __restrict__ embed, const float* __restrict__ WA0,
                            const int* __restrict__ ei, const int* __restrict__ si,
                            float* __restrict__ s1, float* __restrict__ x1)
{
    __shared__ float xs[V_ * H_];     // 20 KB of 320 KB WGP LDS
    const int b = blockIdx.x;
    const int h = threadIdx.x;

    // eA0[h] = embed[1,:] @ WA0[:,h]
    float eA = 0.f;
#pragma unroll
    for (int f = 0; f < FIN_; ++f) eA += embed[FIN_ + f] * WA0[f * H_ + h];

    float gate[NSW_];
#pragma unroll
    for (int k = 0; k < NSW_; ++k) {
        const int ik = si[k], jk = si[NSW_ + k];
        const float e = eA + xB[((size_t)b * V_ + ik) * H_ + h]
                           + xC[((size_t)b * V_ + jk) * H_ + h];
        gate[k] = 1.f / (1.f + expf(-e));
        s1[((size_t)b * NSW_ + k) * H_ + h] = fmaxf(e, 0.f);
    }

    for (int i = 0; i < V_; ++i) xs[i * H_ + h] = Ux[((size_t)b * V_ + i) * H_ + h];
    for (int m = 0; m < NE_; ++m) {
        const int im = ei[m], jm = ei[NE_ + m];
        xs[im * H_ + h] += Vx[((size_t)b * V_ + jm) * H_ + h];
        xs[jm * H_ + h] += Vx[((size_t)b * V_ + im) * H_ + h];
    }
#pragma unroll
    for (int k = 0; k < NSW_; ++k) {
        const int ik = si[k], jk = si[NSW_ + k];
        xs[ik * H_ + h] += gate[k] * Vx[((size_t)b * V_ + jk) * H_ + h];
    }
    for (int i = 0; i < V_; ++i)
        x1[((size_t)b * V_ + i) * H_ + h] = fmaxf(xs[i * H_ + h], 0.f);
}

// ---------------------------------------------------------------------------
// Layer-1 fusion (residual) writing directly into the MLP input buffer:
// mlp_in[b] = [ s2(b,12,128) | x2(b,40,128) ]  (6656 wide)
// ---------------------------------------------------------------------------
__global__ void layer1_post(const float* __restrict__ Ux, const float* __restrict__ Vx,
                            const float* __restrict__ xB, const float* __restrict__ xC,
                            const float* __restrict__ sA1, const float* __restrict__ s1,
                            const float* __restrict__ x1,
                            const int* __restrict__ ei, const int* __restrict__ si,
                            float* __restrict__ mlp_in)
{
    __shared__ float xs[V_ * H_];
    const int b = blockIdx.x;
    const int h = threadIdx.x;
    const size_t mi = (size_t)b * ((NSW_ + V_) * H_);

    float gate[NSW_];
#pragma unroll
    for (int k = 0; k < NSW_; ++k) {
        const int ik = si[k], jk = si[NSW_ + k];
        const float e = sA1[((size_t)b * NSW_ + k) * H_ + h]
                      + xB[((size_t)b * V_ + ik) * H_ + h]
                      + xC[((size_t)b * V_ + jk) * H_ + h];
        gate[k] = 1.f / (1.f + expf(-e));
        // s2 = s1 + relu(e)  -> switches portion of mlp_in
        mlp_in[mi + k * H_ + h] = s1[((size_t)b * NSW_ + k) * H_ + h] + fmaxf(e, 0.f);
    }

    for (int i = 0; i < V_; ++i) xs[i * H_ + h] = Ux[((size_t)b * V_ + i) * H_ + h];
    for (int m = 0; m < NE_; ++m) {
        const int im = ei[m], jm = ei[NE_ + m];
        xs[im * H_ + h] += Vx[((size_t)b * V_ + jm) * H_ + h];
        xs[jm * H_ + h] += Vx[((size_t)b * V_ + im) * H_ + h];
    }
#pragma unroll
    for (int k = 0; k < NSW_; ++k) {
        const int ik = si[k], jk = si[NSW_ + k];
        xs[ik * H_ + h] += gate[k] * Vx[((size_t)b * V_ + jk) * H_ + h];
    }
    for (int i = 0; i < V_; ++i)
        mlp_in[mi + (NSW_ + i) * H_ + h] =
            x1[((size_t)b * V_ + i) * H_ + h] + fmaxf(xs[i * H_ + h], 0.f);
}

// ---------------------------------------------------------------------------
// Final postprocess: per-batch block, builds z (144) and zc (132).
// ---------------------------------------------------------------------------
__global__ void postprocess(const float* __restrict__ outv, const float* __restrict__ A_inc,
                            const float* __restrict__ x_mod,
                            float* __restrict__ z, float* __restrict__ zc)
{
    __shared__ float o[OUT_];
    __shared__ float topo[M_], vv[V_], p_fc[M_], q_fc[M_];
    const int b = blockIdx.x;
    const int t = threadIdx.x;

    if (t < OUT_) o[t] = outv[(size_t)b * OUT_ + t];
    __syncthreads();

    if (t < M_)  topo[t] = (t < M_ - NSW_) ? 1.f
                       : 1.f / (1.f + expf(-o[OUT_ - NSW_ + (t - (M_ - NSW_))]));
    if (t < V_)  vv[t] = (t == 0) ? 1.f : o[M_ + t];
    __syncthreads();

    if (t < M_) {
        p_fc[t] = topo[t] * o[t];
        float a = 0.f;
        for (int n = 0; n < V_; ++n) a += vv[n] * A_inc[n * M_ + t];
        q_fc[t] = topo[t] * a;
    }
    __syncthreads();

    // z = [p_fc(52) | v(40) | graph_topo(52)]
    if (t < M_) z[(size_t)b * 144 + t] = p_fc[t];
    if (t < V_) z[(size_t)b * 144 + M_ + t] = vv[t];
    if (t < M_) z[(size_t)b * 144 + M_ + V_ + t] = topo[t];

    // zc = [q_fc(52) | pg(40) | qg(40)]
    if (t < M_) zc[(size_t)b * 132 + t] = q_fc[t];
    if (t < V_) {
        float pg = x_mod[((size_t)b * V_ + t) * FIN_ + 0];
        float qg = x_mod[((size_t)b * V_ + t) * FIN_ + 1];
        for (int m = 0; m < M_; ++m) {
            pg += A_inc[t * M_ + m] * p_fc[m];
            qg += A_inc[t * M_ + m] * q_fc[m];
        }
        zc[(size_t)b * 132 + M_ + t]      = pg;
        zc[(size_t)b * 132 + M_ + V_ + t] = qg;
    }
}

// ---------------------------------------------------------------------------
extern "C" void kernel_launch(void* const* d_in, const int* in_sizes, int n_in,
                              void* d_out, int out_size, void* d_ws, size_t ws_size,
                              hipStream_t stream)
{
    (void)in_sizes; (void)n_in; (void)out_size; (void)ws_size;

    const float* x_mod = (const float*)d_in[0];   // (200,40,16)
    const int*   ei    = (const int*)  d_in[1];   // (200,2,40) -> use batch 0
    const int*   si    = (const int*)  d_in[2];   // (200,2,12) -> use batch 0
    const float* A_inc = (const float*)d_in[3];   // (40,52)
    const float* embed = (const float*)d_in[4];   // (2,16)
    const float* WU0   = (const float*)d_in[5];
    const float* WV0   = (const float*)d_in[6];
    const float* WA0   = (const float*)d_in[7];
    const float* WB0   = (const float*)d_in[8];
    const float* WC0   = (const float*)d_in[9];
    const float* WU1   = (const float*)d_in[10];
    const float* WV1   = (const float*)d_in[11];
    const float* WA1   = (const float*)d_in[12];
    const float* WB1   = (const float*)d_in[13];
    const float* WC1   = (const float*)d_in[14];
    const float* W1    = (const float*)d_in[15];  // (6656,1024)
    const float* b1    = (const float*)d_in[16];
    const float* W2    = (const float*)d_in[17];  // (1024,104)
    const float* b2    = (const float*)d_in[18];

    float* ws    = (float*)d_ws;
    float* Ux    = ws;                 // 8000*128
    float* Vx    = Ux    + 1024000;    // 8000*128
    float* xB    = Vx    + 1024000;
    float* xC    = xB    + 1024000;
    float* s1    = xC    + 1024000;    // 2400*128
    float* x1    = s1    + 307200;     // 8000*128
    float* sA1   = x1    + 1024000;    // 2400*128
    float* mlpin = sA1   + 307200;     // 200*6656
    float* hid   = mlpin + 1331200;    // 200*1024
    float* outv  = hid   + 204800;     // 200*104

    float* z  = (float*)d_out;                     // (200,144)
    float* zc = (float*)d_out + (size_t)B_ * 144;  // (200,132)

    const dim3 blk(128);
    auto gemm = [&](const float* A, const float* Bm, float* C, const float* bias,
                    int M, int N, int K, int relu) {
        dim3 grd((N + 63) / 64, (M + 63) / 64);
        hipLaunchKernelGGL(wmma_gemm_f32, grd, blk, 0, stream, A, Bm, C, bias, M, N, K, relu);
    };

    // --- Layer 0: x-path projections (8000x16 @ 16x128) ---
    gemm(x_mod, WU0, Ux, nullptr, B_ * V_, H_, FIN_, 0);
    gemm(x_mod, WV0, Vx, nullptr, B_ * V_, H_, FIN_, 0);
    gemm(x_mod, WB0, xB, nullptr, B_ * V_, H_, FIN_, 0);
    gemm(x_mod, WC0, xC, nullptr, B_ * V_, H_, FIN_, 0);
    hipLaunchKernelGGL(layer0_post, dim3(B_), dim3(H_), 0, stream,
                       Ux, Vx, xB, xC, embed, WA0, ei, si, s1, x1);

    // --- Layer 1: x-path projections (8000x128 @ 128x128), s-path (2400x128 @ 128x128) ---
    gemm(x1, WU1, Ux, nullptr, B_ * V_, H_, H_, 0);
    gemm(x1, WV1, Vx, nullptr, B_ * V_, H_, H_, 0);
    gemm(x1, WB1, xB, nullptr, B_ * V_, H_, H_, 0);
    gemm(x1, WC1, xC, nullptr, B_ * V_, H_, H_, 0);
    gemm(s1, WA1, sA1, nullptr, B_ * NSW_, H_, H_, 0);
    hipLaunchKernelGGL(layer1_post, dim3(B_), dim3(H_), 0, stream,
                       Ux, Vx, xB, xC, sA1, s1, x1, ei, si, mlpin);

    // --- MLP: (200x6656 @ 6656x1024) + b1, relu; then (200x1024 @ 1024x104) + b2 ---
    gemm(mlpin, W1, hid, b1, B_, HID_, (NSW_ + V_) * H_, 1);
    gemm(hid,   W2, outv, b2, B_, OUT_, HID_, 0);

    // --- Postprocess -> z, zc ---
    hipLaunchKernelGGL(postprocess, dim3(B_), dim3(128), 0, stream,
                       outv, A_inc, x_mod, z, zc);
}